// RevGNNLayer_21277267984968
// MI455X (gfx1250) — compile-verified
//
#include <hip/hip_runtime.h>

typedef float v2f __attribute__((ext_vector_type(2)));
typedef float v8f __attribute__((ext_vector_type(8)));

#define N_NODES 50000
#define N_EDGES 1600000
#define D 64

// ---------------------------------------------------------------- zero scratch
__global__ void zero_kernel(float* __restrict__ p, int n) {
  int i = blockIdx.x * blockDim.x + threadIdx.x;
  if (i < n) p[i] = 0.0f;
}

// --------------------------------------------- LayerNorm + ReLU + dropout mask
// One wave (32 lanes) per node; each lane owns 2 consecutive features.
__global__ void norm_kernel(const float* __restrict__ x,
                            const float* __restrict__ mask,
                            const float* __restrict__ gamma,
                            const float* __restrict__ beta,
                            float* __restrict__ h) {
  int wave = (blockIdx.x * blockDim.x + threadIdx.x) >> 5;
  int lane = threadIdx.x & 31;
  if (wave >= N_NODES) return;
  const float* row = x + (long)wave * D;
  float2 v = *(const float2*)(row + 2 * lane);

  float s = v.x + v.y;
#pragma unroll
  for (int off = 16; off > 0; off >>= 1) s += __shfl_xor(s, off, 32);
  float mean = s * (1.0f / D);

  float dx = v.x - mean, dy = v.y - mean;
  float q = dx * dx + dy * dy;
#pragma unroll
  for (int off = 16; off > 0; off >>= 1) q += __shfl_xor(q, off, 32);
  float rstd = rsqrtf(q * (1.0f / D) + 1e-5f);

  float2 g = *(const float2*)(gamma + 2 * lane);
  float2 b = *(const float2*)(beta + 2 * lane);
  float2 m = *(const float2*)(mask + (long)wave * D + 2 * lane);
  float2 o;
  o.x = fmaxf(dx * rstd * g.x + b.x, 0.0f) * m.x;
  o.y = fmaxf(dy * rstd * g.y + b.y, 0.0f) * m.y;
  *(float2*)(h + (long)wave * D + 2 * lane) = o;
}

// --------------------------------------------- edge scatter: segment_sum + cnt
// One wave per edge: 32 lanes x float2 = full 64-wide row. Gather of h[src]
// is a coalesced 256B row read (L2-resident: h is 12.8MB << 192MB L2);
// atomic adds land in the L2 atomic units.
__global__ void scatter_kernel(const int* __restrict__ ei,
                               const float* __restrict__ h,
                               float* __restrict__ agg,
                               float* __restrict__ cnt) {
  long t = (long)blockIdx.x * blockDim.x + threadIdx.x;
  int e = (int)(t >> 5);
  int lane = (int)(t & 31);
  if (e >= N_EDGES) return;
  int src = ei[e];
  int dst = ei[N_EDGES + e];
  float2 v = *(const float2*)(h + (long)src * D + 2 * lane);
  float* dp = agg + (long)dst * D + 2 * lane;
  atomicAdd(dp, v.x);
  atomicAdd(dp + 1, v.y);
  if (lane == 0) atomicAdd(cnt + dst, 1.0f);
}

// ------------------------- out = (agg/cnt) @ Wl^T + b_l + h @ Wr^T  via WMMA
// One wave per 16-node tile. fp32 WMMA: V_WMMA_F32_16X16X4_F32.
// A layout: lane holds row M=lane&15; lanes<16 -> K={0,1}, lanes>=16 -> K={2,3}.
// B layout: lane holds col N=lane&15 with same K half-split; B[k][n]=W[n][k]
//           so each B fragment is a contiguous float2 from row n of W.
// C/D: lane holds col n; VGPR v holds row v (lo half) / v+8 (hi half).
__global__ void gemm_kernel(const float* __restrict__ h,
                            const float* __restrict__ agg,
                            const float* __restrict__ cnt,
                            const float* __restrict__ Wl,
                            const float* __restrict__ bl,
                            const float* __restrict__ Wr,
                            float* __restrict__ out) {
  int wave = (blockIdx.x * blockDim.x + threadIdx.x) >> 5;
  int lane = threadIdx.x & 31;
  if (wave >= N_NODES / 16) return;   // uniform per wave: EXEC stays all-1s

  int base = wave * 16;
  int r    = base + (lane & 15);      // node row owned by this lane (A frags)
  int kh   = (lane >> 4) * 2;         // K half: 0 or 2

  float inv = 1.0f / fmaxf(cnt[r], 1.0f);

  // Preload this lane's 32 K-values of agg-row (scaled) and h-row.
  v2f a_agg[16], a_h[16];
  const float* arow = agg + (long)r * D + kh;
  const float* hrow = h   + (long)r * D + kh;
#pragma unroll
  for (int i = 0; i < 16; ++i) {
    float2 ta = *(const float2*)(arow + 4 * i);
    a_agg[i].x = ta.x * inv;
    a_agg[i].y = ta.y * inv;
    float2 th = *(const float2*)(hrow + 4 * i);
    a_h[i].x = th.x;
    a_h[i].y = th.y;
  }

#pragma unroll
  for (int nt = 0; nt < 4; ++nt) {
    int col = nt * 16 + (lane & 15);  // output column owned by this lane
    float cb = bl[col];               // bias: same for every row in the column
    v8f c = {cb, cb, cb, cb, cb, cb, cb, cb};

    const float* wl = Wl + (long)col * D + kh;  // B = Wl^T fragments
    const float* wr = Wr + (long)col * D + kh;  // B = Wr^T fragments
#pragma unroll
    for (int i = 0; i < 16; ++i) {
      float2 tb = *(const float2*)(wl + 4 * i);
      v2f bv; bv.x = tb.x; bv.y = tb.y;
      c = __builtin_amdgcn_wmma_f32_16x16x4_f32(
              false, a_agg[i], false, bv, (short)0, c, false, false);
      float2 tc = *(const float2*)(wr + 4 * i);
      v2f bw; bw.x = tc.x; bw.y = tc.y;
      c = __builtin_amdgcn_wmma_f32_16x16x4_f32(
              false, a_h[i], false, bw, (short)0, c, false, false);
    }

    int hi = (lane >> 4) * 8;
#pragma unroll
    for (int v = 0; v < 8; ++v)
      out[(long)(base + v + hi) * D + col] = c[v];
  }
}

// ---------------------------------------------------------------------- launch
extern "C" void kernel_launch(void* const* d_in, const int* in_sizes, int n_in,
                              void* d_out, int out_size, void* d_ws, size_t ws_size,
                              hipStream_t stream) {
  const float* x     = (const float*)d_in[0];
  const float* mask  = (const float*)d_in[1];
  const float* gamma = (const float*)d_in[2];
  const float* beta  = (const float*)d_in[3];
  const float* Wl    = (const float*)d_in[4];
  const float* bl    = (const float*)d_in[5];
  const float* Wr    = (const float*)d_in[6];
  const int*   ei    = (const int*)d_in[7];
  float* out = (float*)d_out;

  // workspace layout: h[N*D] | agg[N*D] | cnt[N]
  float* h    = (float*)d_ws;
  float* agg  = h + (size_t)N_NODES * D;
  float* cntp = agg + (size_t)N_NODES * D;

  int nz = N_NODES * D + N_NODES;  // agg + cnt are contiguous: zero both
  zero_kernel<<<(nz + 255) / 256, 256, 0, stream>>>(agg, nz);

  norm_kernel<<<(N_NODES + 7) / 8, 256, 0, stream>>>(x, mask, gamma, beta, h);

  long nthreads = (long)N_EDGES * 32;
  scatter_kernel<<<(int)((nthreads + 255) / 256), 256, 0, stream>>>(ei, h, agg, cntp);

  int tiles = N_NODES / 16;  // 3125, exact
  gemm_kernel<<<(tiles + 3) / 4, 128, 0, stream>>>(h, agg, cntp, Wl, bl, Wr, out);
}